// EvolutionCrossAttention_75539884802647
// MI455X (gfx1250) — compile-verified
//
#include <hip/hip_runtime.h>
#include <math.h>

typedef __attribute__((ext_vector_type(2))) float v2f;
typedef __attribute__((ext_vector_type(8))) float v8f;

#define B_   4
#define C_   128
#define N_   131072     /* 32*64*64 */
#define E_   128
#define H_   4
#define HD_  32
#define G_   8
#define CG_  16
#define EPSV 1e-5f

/* workspace layout (float offsets) */
#define WS_STATS   0        /* 64  : (sum,sumsq) per (b,g)           */
#define WS_MU      64       /* 32                                    */
#define WS_RS      96       /* 32                                    */
#define WS_SCONST  128      /* 16                                    */
#define WS_Z       144      /* 16  : softmax denominators            */
#define WS_MAX     160      /* 16  : ordered-uint maxima             */
#define WS_T       176      /* 2048: raw weighted sums t[b][h][c]    */
#define WS_W2      2224     /* 8192: folded A matrix [b][c][row16]   */
#define WS_SBUF    10416    /* 2097152: scores/probs [b][h][n]       */

__device__ __forceinline__ unsigned f2o(float f) {
    unsigned u = __float_as_uint(f);
    return (u & 0x80000000u) ? ~u : (u | 0x80000000u);
}
__device__ __forceinline__ float o2f(unsigned u) {
    return __uint_as_float((u & 0x80000000u) ? (u & 0x7fffffffu) : ~u);
}
__device__ __forceinline__ float wave_sum(float v) {
    for (int o = 16; o > 0; o >>= 1) v += __shfl_xor(v, o, 32);
    return v;
}
__device__ __forceinline__ float wave_max(float v) {
    for (int o = 16; o > 0; o >>= 1) v = fmaxf(v, __shfl_xor(v, o, 32));
    return v;
}

/* ---------------- K0: zero the small accumulator regions ---------------- */
__global__ void k_init(float* ws) {
    int t = threadIdx.x;
    for (int i = t; i < 64 + 2048; i += 256) {
        if (i < 64) ws[WS_STATS + i] = 0.f;
        else        ws[WS_T + (i - 64)] = 0.f;
    }
    if (t < 16) { ws[WS_Z + t] = 0.f; ((unsigned*)ws)[WS_MAX + t] = 0u; }
}

/* ---------------- K1: GroupNorm statistics (sum, sumsq) ----------------- */
/* grid (128, 8, 4), block 256 : slice of 1024 positions x 16 channels     */
__global__ void k_gnstats(const float* __restrict__ x, float* __restrict__ ws) {
    const int b = blockIdx.z, g = blockIdx.y;
    const int n0 = blockIdx.x * 1024;
    const int t = threadIdx.x;
    float s = 0.f, ss = 0.f;
    for (int cc = 0; cc < CG_; ++cc) {
        const float* p = x + ((b * C_ + g * CG_ + cc) * (long)N_) + n0;
        for (int k = t; k < 1024; k += 256) {
            float v = p[k];
            s += v; ss += v * v;
        }
    }
    __shared__ float r1[8], r2[8];
    s = wave_sum(s); ss = wave_sum(ss);
    int lane = t & 31, w = t >> 5;
    if (lane == 0) { r1[w] = s; r2[w] = ss; }
    __syncthreads();
    if (t == 0) {
        float a = 0.f, bb = 0.f;
        for (int i = 0; i < 8; ++i) { a += r1[i]; bb += r2[i]; }
        atomicAdd(&ws[WS_STATS + 2 * (b * G_ + g)], a);
        atomicAdd(&ws[WS_STATS + 2 * (b * G_ + g) + 1], bb);
    }
}

/* ---------------- K2: LN, q, folded score weights (tiny, 1 block) ------- */
__global__ void k_prep(const float* __restrict__ ef,
                       const float* __restrict__ ln_g, const float* __restrict__ ln_b,
                       const float* __restrict__ gn_g, const float* __restrict__ gn_b,
                       const float* __restrict__ Wq, const float* __restrict__ bq,
                       const float* __restrict__ Wk, const float* __restrict__ bk,
                       float* __restrict__ ws) {
    const int c = threadIdx.x;                 /* 128 threads */
    __shared__ float e_lds[B_ * E_];
    __shared__ float q_lds[B_ * C_];
    __shared__ float wv_lds[16 * C_];
    __shared__ float mu_l[32], rs_l[32];
    const float scale = rsqrtf((float)HD_);

    for (int b = 0; b < B_; ++b) e_lds[b * E_ + c] = ef[b * E_ + c];
    __syncthreads();

    /* q = LN(e) @ Wq^T + bq  (redundant per-thread LN stats, trivial) */
    for (int b = 0; b < B_; ++b) {
        float mu = 0.f;
        for (int j = 0; j < E_; ++j) mu += e_lds[b * E_ + j];
        mu *= (1.f / E_);
        float var = 0.f;
        for (int j = 0; j < E_; ++j) { float d = e_lds[b * E_ + j] - mu; var += d * d; }
        var *= (1.f / E_);
        float rs = rsqrtf(var + EPSV);
        float q = bq[c];
        for (int j = 0; j < E_; ++j) {
            float eln = (e_lds[b * E_ + j] - mu) * rs * ln_g[j] + ln_b[j];
            q += eln * Wq[c * E_ + j];
        }
        q_lds[b * C_ + c] = q;
    }
    __syncthreads();

    /* GN stats -> mu, rs per (b,g) */
    if (c < 32) {
        float cnt = (float)CG_ * (float)N_;
        float s  = ws[WS_STATS + 2 * c];
        float ss = ws[WS_STATS + 2 * c + 1];
        float mu = s / cnt;
        float var = ss / cnt - mu * mu;
        float rs = rsqrtf(var + EPSV);
        mu_l[c] = mu; rs_l[c] = rs;
        ws[WS_MU + c] = mu; ws[WS_RS + c] = rs;
    }
    __syncthreads();

    /* wvec[b][h][c] = scale * sum_d q_d Wk[h*32+d, c] */
    for (int b = 0; b < B_; ++b)
        for (int h = 0; h < H_; ++h) {
            float wv = 0.f;
            for (int d = 0; d < HD_; ++d)
                wv += q_lds[b * C_ + h * HD_ + d] * Wk[(h * HD_ + d) * C_ + c];
            wv_lds[(b * H_ + h) * C_ + c] = wv * scale;
        }
    __syncthreads();

    /* sconst[b][h] = scale*q·bk + sum_c wvec_c*(gn_b - mu*rs*gn_g) */
    if (c < 16) {
        int b = c >> 2, h = c & 3;
        float sc = 0.f;
        for (int d = 0; d < HD_; ++d)
            sc += q_lds[b * C_ + h * HD_ + d] * bk[h * HD_ + d];
        sc *= scale;
        for (int j = 0; j < C_; ++j) {
            int g = j >> 4;
            float term = gn_b[j] - mu_l[b * G_ + g] * rs_l[b * G_ + g] * gn_g[j];
            sc += wv_lds[c * C_ + j] * term;
        }
        ws[WS_SCONST + c] = sc;
    }

    /* folded A matrix: w2[b][c][row], rows 4..15 zero */
    for (int b = 0; b < B_; ++b) {
        int g = c >> 4;
        float f = gn_g[c] * rs_l[b * G_ + g];
        for (int h = 0; h < H_; ++h)
            ws[WS_W2 + b * 2048 + c * 16 + h] = wv_lds[(b * H_ + h) * C_ + c] * f;
        for (int r = H_; r < 16; ++r)
            ws[WS_W2 + b * 2048 + c * 16 + r] = 0.f;
    }
}

/* ---------------- K3: WMMA score pass  s = A(16x128) @ X(128xN) --------- */
/* grid (64, B), block 256 (8 waves); 16 tiles of 16 columns per wave      */
__global__ void k_scores(const float* __restrict__ x, float* __restrict__ ws) {
    __shared__ float a_lds[C_ * 16];     /* [c][row] */
    __shared__ float sc_lds[16];
    const int b = blockIdx.y;
    const int tid = threadIdx.x;
    for (int i = tid; i < C_ * 16; i += 256) a_lds[i] = ws[WS_W2 + b * 2048 + i];
    if (tid < 4) sc_lds[tid] = ws[WS_SCONST + b * 4 + tid];
    __syncthreads();

    const int lane = tid & 31, wave = tid >> 5;
    const int row = lane & 15;           /* A row / B column within tile */
    const int khalf = lane >> 4;         /* K pair selector              */
    const float* xb = x + (long)b * C_ * N_;
    float* sbuf = ws + WS_SBUF;
    unsigned* maxbuf = (unsigned*)ws + WS_MAX;

    float ml[4] = { -3.4e38f, -3.4e38f, -3.4e38f, -3.4e38f };

    const int tile0 = blockIdx.x * 128 + wave;
    for (int i = 0; i < 16; ++i) {
        const int n0 = (tile0 + i * 8) * 16;
        v8f acc = {};
#pragma unroll 8
        for (int c0 = 0; c0 < C_; c0 += 4) {
            const int cA = c0 + khalf * 2;
            v2f a, bm;
            a.x = a_lds[cA * 16 + row];
            a.y = a_lds[(cA + 1) * 16 + row];
            bm.x = xb[cA * N_ + n0 + row];
            bm.y = xb[(cA + 1) * N_ + n0 + row];
            acc = __builtin_amdgcn_wmma_f32_16x16x4_f32(
                false, a, false, bm, (short)0, acc, false, false);
        }
        if (lane < 16) {
#pragma unroll
            for (int h = 0; h < H_; ++h) {
                float sv = acc[h] + sc_lds[h];
                sbuf[((b * H_ + h) << 17) + n0 + lane] = sv;
                ml[h] = fmaxf(ml[h], sv);
            }
        }
    }
#pragma unroll
    for (int h = 0; h < H_; ++h) {
        float m = wave_max(ml[h]);
        if (lane == 0) atomicMax(&maxbuf[b * H_ + h], f2o(m));
    }
}

/* ---------------- K4: p = exp(s - max); accumulate Z -------------------- */
/* grid (64, 16), block 256 : 2048 elements per block within one (b,h)     */
__global__ void k_softexp(float* __restrict__ ws) {
    const int bh = blockIdx.y;
    const float m = o2f(((unsigned*)ws)[WS_MAX + bh]);
    float* s = ws + WS_SBUF + ((long)bh << 17) + blockIdx.x * 2048;
    const int t = threadIdx.x;
    float z = 0.f;
#pragma unroll
    for (int i = 0; i < 8; ++i) {
        float p = __expf(s[i * 256 + t] - m);
        s[i * 256 + t] = p;
        z += p;
    }
    __shared__ float r[8];
    z = wave_sum(z);
    if ((t & 31) == 0) r[t >> 5] = z;
    __syncthreads();
    if (t == 0) {
        float tot = 0.f;
        for (int i = 0; i < 8; ++i) tot += r[i];
        atomicAdd(&ws[WS_Z + bh], tot);
    }
}

/* ---------------- K5: t[b][h][c] = sum_n p[b][h][n] * x[b][c][n] -------- */
/* grid (128, B), block 256 (8 waves x 16 channels); float4 loads          */
__global__ void k_accum(const float* __restrict__ x, float* __restrict__ ws) {
    const int b = blockIdx.y;
    const int tid = threadIdx.x, lane = tid & 31, wave = tid >> 5;
    const int cbase = wave * 16;
    const int n0 = blockIdx.x * 1024;
    const float* xb = x + (long)b * C_ * N_;
    const float* pb = ws + WS_SBUF + ((long)b * H_ << 17);

    float acc[16][4];
#pragma unroll
    for (int cc = 0; cc < 16; ++cc)
#pragma unroll
        for (int h = 0; h < 4; ++h) acc[cc][h] = 0.f;

    for (int i = 0; i < 8; ++i) {
        const int n = n0 + i * 128 + lane * 4;
        float4 p4[4];
#pragma unroll
        for (int h = 0; h < 4; ++h)
            p4[h] = *(const float4*)(pb + ((long)h << 17) + n);
#pragma unroll
        for (int cc = 0; cc < 16; ++cc) {
            float4 xv = *(const float4*)(xb + (long)(cbase + cc) * N_ + n);
#pragma unroll
            for (int h = 0; h < 4; ++h)
                acc[cc][h] += xv.x * p4[h].x + xv.y * p4[h].y +
                              xv.z * p4[h].z + xv.w * p4[h].w;
        }
    }
#pragma unroll
    for (int cc = 0; cc < 16; ++cc)
#pragma unroll
        for (int h = 0; h < 4; ++h) {
            float v = wave_sum(acc[cc][h]);
            if (lane == 0)
                atomicAdd(&ws[WS_T + (b * H_ + h) * C_ + cbase + cc], v);
        }
}

/* ---------------- K6: epilogue  Wv, Wo  (tiny, 1 block) ----------------- */
__global__ void k_final(const float* __restrict__ gn_g, const float* __restrict__ gn_b,
                        const float* __restrict__ Wv, const float* __restrict__ bv,
                        const float* __restrict__ Wo, const float* __restrict__ bo,
                        const float* __restrict__ ws, float* __restrict__ out) {
    const int c = threadIdx.x;           /* 128 threads */
    __shared__ float xt[16 * C_];
    __shared__ float ov[B_ * C_];
    for (int bh = 0; bh < 16; ++bh) {
        int b = bh >> 2, g = c >> 4;
        float mu = ws[WS_MU + b * G_ + g], rs = ws[WS_RS + b * G_ + g];
        float t = ws[WS_T + bh * C_ + c] / ws[WS_Z + bh];
        xt[bh * C_ + c] = gn_g[c] * rs * t + (gn_b[c] - mu * rs * gn_g[c]);
    }
    __syncthreads();
    for (int b = 0; b < B_; ++b) {
        int h = c >> 5;
        float o = bv[c];
        for (int j = 0; j < C_; ++j)
            o += Wv[c * C_ + j] * xt[(b * H_ + h) * C_ + j];
        ov[b * C_ + c] = o;
    }
    __syncthreads();
    for (int b = 0; b < B_; ++b) {
        float o = bo[c];
        for (int j = 0; j < C_; ++j)
            o += Wo[c * C_ + j] * ov[b * C_ + j];
        out[b * C_ + c] = o;
    }
}

extern "C" void kernel_launch(void* const* d_in, const int* in_sizes, int n_in,
                              void* d_out, int out_size, void* d_ws, size_t ws_size,
                              hipStream_t stream) {
    const float* x    = (const float*)d_in[0];
    const float* ef   = (const float*)d_in[1];
    const float* ln_g = (const float*)d_in[2];
    const float* ln_b = (const float*)d_in[3];
    const float* gn_g = (const float*)d_in[4];
    const float* gn_b = (const float*)d_in[5];
    const float* Wq   = (const float*)d_in[6];
    const float* bq   = (const float*)d_in[7];
    const float* Wk   = (const float*)d_in[8];
    const float* bk   = (const float*)d_in[9];
    const float* Wv   = (const float*)d_in[10];
    const float* bv   = (const float*)d_in[11];
    const float* Wo   = (const float*)d_in[12];
    const float* bo   = (const float*)d_in[13];
    float* ws  = (float*)d_ws;
    float* out = (float*)d_out;

    k_init<<<1, 256, 0, stream>>>(ws);
    k_gnstats<<<dim3(128, G_, B_), 256, 0, stream>>>(x, ws);
    k_prep<<<1, 128, 0, stream>>>(ef, ln_g, ln_b, gn_g, gn_b, Wq, bq, Wk, bk, ws);
    k_scores<<<dim3(64, B_), 256, 0, stream>>>(x, ws);
    k_softexp<<<dim3(64, 16), 256, 0, stream>>>(ws);
    k_accum<<<dim3(128, B_), 256, 0, stream>>>(x, ws);
    k_final<<<1, 128, 0, stream>>>(gn_g, gn_b, Wv, bv, Wo, bo, ws, out);
}